// Decoder_24601572671811
// MI455X (gfx1250) — compile-verified
//
#include <hip/hip_runtime.h>

typedef __attribute__((ext_vector_type(2))) float v2f;
typedef __attribute__((ext_vector_type(8))) float v8f;

#define Bdim 4
#define Cdim 2
#define Edim 256
#define Ldim 16000
#define Wdim 40
#define STEP 20
#define SEGS (Ldim + 1)                                   // 16001 output segments
#define SEG_PER_TILE 15
#define NTILES ((SEGS + SEG_PER_TILE - 1) / SEG_PER_TILE) // 1067
#define T_OUT (SEGS * STEP)                               // 320020
#define WAVES_PER_BLOCK 4
#define NPAD 48            // N padded to 3 WMMA tiles
#define BSTRIDE 258        // padded LDS row stride (dwords): banks 2n+k -> conflict-free

// One wave = one 16-row M-tile of the (16000 x 40) = (L x W) GEMM for one (b,c),
// contracting K=E=256 with V_WMMA_F32_16X16X4_F32. Basis is staged zero-padded
// into LDS once per block so the inner loop is branch-free. Each wave then does
// the overlap-and-add for its 15 exclusively-owned output segments via LDS.
__global__ __launch_bounds__(WAVES_PER_BLOCK * 32)
void decoder_oa_wmma(const float* __restrict__ mix,   // [B,E,L]
                     const float* __restrict__ mask,  // [B,C,E,L]
                     const float* __restrict__ basis, // [W,E]
                     float* __restrict__ out)         // [B,C,T_OUT]
{
    __shared__ float ldsB[NPAD * BSTRIDE];            // zero-padded basis [48][258]
    __shared__ float ldsD[WAVES_PER_BLOCK][16][48];   // per-wave D staging

    const int tid = threadIdx.x;

    // ---- stage zero-padded basis into LDS (once per block) --------------
    for (int idx = tid; idx < NPAD * Edim; idx += WAVES_PER_BLOCK * 32) {
        const int n = idx >> 8;        // idx / 256
        const int k = idx & 255;       // idx % 256
        ldsB[n * BSTRIDE + k] = (n < Wdim) ? basis[n * Edim + k] : 0.f;
    }
    __syncthreads();

    const int lane    = tid & 31;
    const int wv      = tid >> 5;
    int wave_id       = blockIdx.x * WAVES_PER_BLOCK + wv;
    const int total   = Bdim * Cdim * NTILES;
    if (wave_id >= total) wave_id = total - 1;   // duplicates write identical values

    const int bc   = wave_id / NTILES;
    const int tile = wave_id % NTILES;
    const int b    = bc / Cdim;
    const int c    = bc % Cdim;
    const int r0   = tile * SEG_PER_TILE - 1;    // first frames row (may be -1)

    // --- per-lane WMMA operand coordinates -------------------------------
    const int halfsel = lane >> 4;               // 0: K pair {0,1}, 1: K pair {2,3}
    const int ncol    = lane & 15;               // A: row within tile / B: column
    const int kofs    = halfsel * 2;
    int mrow = r0 + ncol;                        // frames row for A
    mrow = mrow < 0 ? 0 : (mrow > Ldim - 1 ? Ldim - 1 : mrow);  // clamps never read

    const float* mixp  = mix  + (size_t)b * Edim * Ldim + mrow;
    const float* maskp = mask + (size_t)(b * Cdim + c) * Edim * Ldim + mrow;

    const int bofs0 = (ncol     ) * BSTRIDE;     // B-tile column bases in LDS
    const int bofs1 = (ncol + 16) * BSTRIDE;
    const int bofs2 = (ncol + 32) * BSTRIDE;

    v8f acc0 = {}; v8f acc1 = {}; v8f acc2 = {};

    for (int kk = 0; kk < Edim; kk += 4) {
        const int k0 = kk + kofs;                // even -> 8B-aligned ds_load_b64
        // A[m][k0], A[m][k0+1]  with A = mixture * mask (elementwise over E,L)
        float m0 = mixp [(size_t)k0 * Ldim];
        float m1 = mixp [(size_t)(k0 + 1) * Ldim];
        float s0 = maskp[(size_t)k0 * Ldim];
        float s1 = maskp[(size_t)(k0 + 1) * Ldim];
        v2f a; a.x = m0 * s0; a.y = m1 * s1;

        // B[k][n] = padded basis from LDS; branch-free, bank-conflict-free
        v2f b0 = *(const v2f*)&ldsB[bofs0 + k0];
        v2f b1 = *(const v2f*)&ldsB[bofs1 + k0];
        v2f b2 = *(const v2f*)&ldsB[bofs2 + k0];

        acc0 = __builtin_amdgcn_wmma_f32_16x16x4_f32(false, a, false, b0, (short)0, acc0, false, false);
        acc1 = __builtin_amdgcn_wmma_f32_16x16x4_f32(false, a, false, b1, (short)0, acc1, false, false);
        acc2 = __builtin_amdgcn_wmma_f32_16x16x4_f32(false, a, false, b2, (short)0, acc2, false, false);
    }

    // --- spill D tiles (16x48) to LDS in WMMA C/D layout -----------------
    // VGPR v, lane L: row = v + 8*(L>=16), col = L&15
    const int drow = halfsel * 8;
    #pragma unroll
    for (int v = 0; v < 8; ++v) {
        ldsD[wv][drow + v][ncol     ] = acc0[v];
        ldsD[wv][drow + v][ncol + 16] = acc1[v];
        ldsD[wv][drow + v][ncol + 32] = acc2[v];
    }
    __syncthreads();   // uniform control flow; orders DS stores vs. DS loads

    // --- overlap-and-add: out[20s+j] = frames[s][j] + frames[s-1][j+20] ---
    // This tile owns segments s in [15*tile, 15*tile+15); LDS row of frames
    // row r is (r - r0), so term1 row = sl+1, term2 row = sl.
    float* outp = out + (size_t)bc * T_OUT;
    for (int idx = lane; idx < SEG_PER_TILE * STEP; idx += 32) {
        const int sl = idx / STEP;
        const int j  = idx % STEP;
        const int s  = tile * SEG_PER_TILE + sl;
        if (s <= Ldim) {
            float v1 = (s < Ldim) ? ldsD[wv][sl + 1][j]        : 0.f;  // frames[s][j]
            float v2 = (s >= 1)   ? ldsD[wv][sl    ][j + STEP] : 0.f;  // frames[s-1][j+20]
            outp[(size_t)s * STEP + j] = v1 + v2;
        }
    }
}

extern "C" void kernel_launch(void* const* d_in, const int* in_sizes, int n_in,
                              void* d_out, int out_size, void* d_ws, size_t ws_size,
                              hipStream_t stream) {
    (void)in_sizes; (void)n_in; (void)out_size; (void)d_ws; (void)ws_size;
    const float* mix   = (const float*)d_in[0];  // mixture_w [4,256,16000]
    const float* mask  = (const float*)d_in[1];  // est_mask  [4,2,256,16000]
    const float* basis = (const float*)d_in[2];  // basis_w   [40,256]
    float* out = (float*)d_out;                  // [4,2,320020]

    const int total_waves = Bdim * Cdim * NTILES;                 // 8536
    const int grid = (total_waves + WAVES_PER_BLOCK - 1) / WAVES_PER_BLOCK;
    decoder_oa_wmma<<<grid, WAVES_PER_BLOCK * 32, 0, stream>>>(mix, mask, basis, out);
}